// OutputLayer_54485955117837
// MI455X (gfx1250) — compile-verified
//
#include <hip/hip_runtime.h>
#include <stdint.h>

// Problem constants (fixed by the reference).
#define TQ 64
#define TP 2048
#define BB 64
#define HH 256
#define AA 75
#define NPAD 80          // A padded to 5 x 16 WMMA tiles
#define NEG_INF (-1e12f)

typedef float v2f __attribute__((ext_vector_type(2)));
typedef float v8f __attribute__((ext_vector_type(8)));

// ---------------------------------------------------------------------------
// GEMM: Y[m, n] = sum_k X[m, k] * W[n, k]   (n < NPAD, rows n >= NW are zero)
// X: [M, ldX] row-major, W: [NW, ldW] row-major, Y: [M, NPAD] row-major.
// K multiple of 4 (here 256). Uses V_WMMA_F32_16X16X4_F32 (fp32 precision).
// Block: 256 threads = 8 waves. Each wave computes TWO 16-row x 80-col strips
// (rows m0..m0+31) so that every B-fragment loaded from LDS feeds 2 WMMAs —
// 10 wmma : 5 ds-loads : 2 global loads per K-step.
// ---------------------------------------------------------------------------
__global__ __launch_bounds__(256) void gemm_xwT_kernel(
    const float* __restrict__ X, const float* __restrict__ W,
    float* __restrict__ Y, int M, int K, int ldX, int ldW, int NW) {
  // LDS tile of W: [NPAD rows][K cols], padded stride to avoid bank conflicts.
  __shared__ float ldsW[NPAD * 260];

  const int tid = threadIdx.x;
  // Cooperative fill: NPAD*K elements, zero-fill rows >= NW.
  for (int idx = tid; idx < NPAD * HH; idx += 256) {
    const int n = idx / HH;
    const int k = idx - n * HH;
    float v = 0.0f;
    if (n < NW) v = W[(size_t)n * ldW + k];
    ldsW[n * 260 + k] = v;
  }
  __syncthreads();

  const int lane = tid & 31;
  const int wave = tid >> 5;
  const int m0   = blockIdx.x * 256 + wave * 32;  // first of two 16-row tiles
  const int ml   = lane & 15;      // M (for A) / N (for B,C,D) sub-index
  const int half = lane >> 4;      // 0: K 0..1 | 1: K 2..3 (for A/B frags)

  // Clamped source rows (guarantee in-bounds loads; stores are guarded).
  int rm0 = m0 + ml;
  int rm1 = m0 + 16 + ml;
  if (rm0 > M - 1) rm0 = M - 1;
  if (rm1 > M - 1) rm1 = M - 1;
  const float* xrow0 = X + (size_t)rm0 * ldX;
  const float* xrow1 = X + (size_t)rm1 * ldX;

  v8f c00 = {}, c01 = {}, c02 = {}, c03 = {}, c04 = {};  // tile 0 (rows m0..)
  v8f c10 = {}, c11 = {}, c12 = {}, c13 = {}, c14 = {};  // tile 1 (rows m0+16..)

  for (int k = 0; k < K; k += 4) {
    const v2f a0 = *reinterpret_cast<const v2f*>(xrow0 + k + half * 2);
    const v2f a1 = *reinterpret_cast<const v2f*>(xrow1 + k + half * 2);
    const v2f b0 = *reinterpret_cast<const v2f*>(&ldsW[(0 * 16 + ml) * 260 + k + half * 2]);
    const v2f b1 = *reinterpret_cast<const v2f*>(&ldsW[(1 * 16 + ml) * 260 + k + half * 2]);
    const v2f b2 = *reinterpret_cast<const v2f*>(&ldsW[(2 * 16 + ml) * 260 + k + half * 2]);
    const v2f b3 = *reinterpret_cast<const v2f*>(&ldsW[(3 * 16 + ml) * 260 + k + half * 2]);
    const v2f b4 = *reinterpret_cast<const v2f*>(&ldsW[(4 * 16 + ml) * 260 + k + half * 2]);
    c00 = __builtin_amdgcn_wmma_f32_16x16x4_f32(false, a0, false, b0, (short)0, c00, false, false);
    c10 = __builtin_amdgcn_wmma_f32_16x16x4_f32(false, a1, false, b0, (short)0, c10, false, false);
    c01 = __builtin_amdgcn_wmma_f32_16x16x4_f32(false, a0, false, b1, (short)0, c01, false, false);
    c11 = __builtin_amdgcn_wmma_f32_16x16x4_f32(false, a1, false, b1, (short)0, c11, false, false);
    c02 = __builtin_amdgcn_wmma_f32_16x16x4_f32(false, a0, false, b2, (short)0, c02, false, false);
    c12 = __builtin_amdgcn_wmma_f32_16x16x4_f32(false, a1, false, b2, (short)0, c12, false, false);
    c03 = __builtin_amdgcn_wmma_f32_16x16x4_f32(false, a0, false, b3, (short)0, c03, false, false);
    c13 = __builtin_amdgcn_wmma_f32_16x16x4_f32(false, a1, false, b3, (short)0, c13, false, false);
    c04 = __builtin_amdgcn_wmma_f32_16x16x4_f32(false, a0, false, b4, (short)0, c04, false, false);
    c14 = __builtin_amdgcn_wmma_f32_16x16x4_f32(false, a1, false, b4, (short)0, c14, false, false);
  }

  // C/D layout: VGPR v, lanes 0-15 -> M=v, lanes 16-31 -> M=v+8; N = lane&15.
  v8f acc0[5] = {c00, c01, c02, c03, c04};
  v8f acc1[5] = {c10, c11, c12, c13, c14};
#pragma unroll
  for (int j = 0; j < 5; ++j) {
#pragma unroll
    for (int v = 0; v < 8; ++v) {
      const int r0 = m0 + half * 8 + v;
      const int r1 = r0 + 16;
      if (r0 < M) Y[(size_t)r0 * NPAD + j * 16 + ml] = acc0[j][v];
      if (r1 < M) Y[(size_t)r1 * NPAD + j * 16 + ml] = acc1[j][v];
    }
  }
}

// ---------------------------------------------------------------------------
// logits[t,b] = mask ? sum_a tanh(Y[t*B+b, a] + spart[b*ss + a]) * w2[a] : -INF
// Optionally also writes the transposed masked logits (final output layout).
// ---------------------------------------------------------------------------
__global__ __launch_bounds__(256) void logits_mask_kernel(
    const float* __restrict__ Y, const float* __restrict__ spart, int sstride,
    const float* __restrict__ w2, const uint8_t* __restrict__ mask,
    int T, float* __restrict__ logits_out, float* __restrict__ trans_out) {
  const int idx = blockIdx.x * 256 + threadIdx.x;  // token index t*B + b
  if (idx >= T * BB) return;
  const int b = idx & (BB - 1);
  const int t = idx >> 6;  // BB == 64
  const float* yrow = Y + (size_t)idx * NPAD;
  const float* srow = spart + (size_t)b * sstride;
  float acc = 0.0f;
  for (int a = 0; a < AA; ++a) acc += tanhf(yrow[a] + srow[a]) * w2[a];
  const float l = mask[idx] ? acc : NEG_INF;
  if (logits_out) logits_out[idx] = l;
  if (trans_out) trans_out[(size_t)b * T + t] = l;
}

// ---------------------------------------------------------------------------
// Masked softmax over t (masked entries carry -1e12 already) + weighted pool:
// pooled[b, h] = sum_t softmax(logits[:, b])[t] * X[t, b, h]
// One block per batch; 256 threads (== H).
// ---------------------------------------------------------------------------
__global__ __launch_bounds__(256) void softmax_pool_kernel(
    const float* __restrict__ logits, const float* __restrict__ X,
    int T, float* __restrict__ pooled) {
  __shared__ float red[256];
  __shared__ float wbuf[TP];  // softmax weights (T <= TP)
  const int b = blockIdx.x;
  const int tid = threadIdx.x;

  // max over t
  float mx = NEG_INF;
  for (int t = tid; t < T; t += 256) mx = fmaxf(mx, logits[(size_t)t * BB + b]);
  red[tid] = mx;
  __syncthreads();
  for (int s = 128; s > 0; s >>= 1) {
    if (tid < s) red[tid] = fmaxf(red[tid], red[tid + s]);
    __syncthreads();
  }
  mx = red[0];
  __syncthreads();

  // sum of exp
  float sm = 0.0f;
  for (int t = tid; t < T; t += 256) sm += __expf(logits[(size_t)t * BB + b] - mx);
  red[tid] = sm;
  __syncthreads();
  for (int s = 128; s > 0; s >>= 1) {
    if (tid < s) red[tid] += red[tid + s];
    __syncthreads();
  }
  sm = red[0];
  __syncthreads();

  const float inv = 1.0f / sm;
  for (int t = tid; t < T; t += 256)
    wbuf[t] = __expf(logits[(size_t)t * BB + b] - mx) * inv;
  __syncthreads();

  // pooled over h = tid
  float acc = 0.0f;
  const float* xb = X + (size_t)b * HH + tid;
  for (int t = 0; t < T; ++t) acc += wbuf[t] * xb[(size_t)t * BB * HH];
  pooled[(size_t)b * HH + tid] = acc;
}

// ---------------------------------------------------------------------------
// GRU cell (torch semantics): one block per batch, one thread per h.
// ---------------------------------------------------------------------------
__global__ __launch_bounds__(256) void gru_kernel(
    const float* __restrict__ x, const float* __restrict__ hprev,
    const float* __restrict__ W_ih, const float* __restrict__ W_hh,
    const float* __restrict__ b_ih, const float* __restrict__ b_hh,
    float* __restrict__ hout) {
  __shared__ float sx[HH];
  __shared__ float sh[HH];
  const int b = blockIdx.x;
  const int h = threadIdx.x;
  sx[h] = x[(size_t)b * HH + h];
  sh[h] = hprev[(size_t)b * HH + h];
  __syncthreads();

  float ir = b_ih[h], iz = b_ih[HH + h], in_ = b_ih[2 * HH + h];
  float hr = b_hh[h], hz = b_hh[HH + h], hn = b_hh[2 * HH + h];
  const float* wr_i = W_ih + (size_t)h * HH;
  const float* wz_i = W_ih + (size_t)(HH + h) * HH;
  const float* wn_i = W_ih + (size_t)(2 * HH + h) * HH;
  const float* wr_h = W_hh + (size_t)h * HH;
  const float* wz_h = W_hh + (size_t)(HH + h) * HH;
  const float* wn_h = W_hh + (size_t)(2 * HH + h) * HH;
  for (int k = 0; k < HH; ++k) {
    const float xv = sx[k], hv = sh[k];
    ir += xv * wr_i[k];  iz += xv * wz_i[k];  in_ += xv * wn_i[k];
    hr += hv * wr_h[k];  hz += hv * wz_h[k];  hn += hv * wn_h[k];
  }
  const float r = 1.0f / (1.0f + __expf(-(ir + hr)));
  const float z = 1.0f / (1.0f + __expf(-(iz + hz)));
  const float n = tanhf(in_ + r * hn);
  hout[(size_t)b * HH + h] = (1.0f - z) * n + z * sh[h];
}

// ---------------------------------------------------------------------------
extern "C" void kernel_launch(void* const* d_in, const int* in_sizes, int n_in,
                              void* d_out, int out_size, void* d_ws, size_t ws_size,
                              hipStream_t stream) {
  (void)in_sizes; (void)n_in; (void)out_size; (void)ws_size;
  const float*   question = (const float*)d_in[0];   // [TQ,BB,HH]
  const float*   passage  = (const float*)d_in[1];   // [TP,BB,HH]
  const float*   V_q      = (const float*)d_in[2];   // [HH]
  const float*   W1_q     = (const float*)d_in[3];   // [AA, 2*HH]
  const float*   W2_q     = (const float*)d_in[4];   // [1, AA]
  const float*   W1_p     = (const float*)d_in[5];   // [AA, 2*HH]
  const float*   W2_p     = (const float*)d_in[6];   // [1, AA]
  const float*   W_ih     = (const float*)d_in[7];   // [3*HH, HH]
  const float*   W_hh     = (const float*)d_in[8];   // [3*HH, HH]
  const float*   b_ih     = (const float*)d_in[9];   // [3*HH]
  const float*   b_hh     = (const float*)d_in[10];  // [3*HH]
  const uint8_t* qmask    = (const uint8_t*)d_in[11]; // [TQ,BB] bool
  const uint8_t* pmask    = (const uint8_t*)d_in[12]; // [TP,BB] bool
  float* out = (float*)d_out;                         // [2, BB, TP]

  // Workspace layout (floats).
  float* ws = (float*)d_ws;
  size_t o = 0;
  float* Pp   = ws + o; o += (size_t)TP * BB * NPAD;  // passage @ W1_p[:, :H].T
  float* Yq   = ws + o; o += (size_t)TQ * BB * NPAD;  // question @ W1_q[:, :H].T
  float* qlog = ws + o; o += (size_t)TQ * BB;
  float* plog = ws + o; o += (size_t)TP * BB;
  float* st1  = ws + o; o += (size_t)BB * HH;         // state after question pool
  float* ci   = ws + o; o += (size_t)BB * HH;         // cell_input
  float* st2  = ws + o; o += (size_t)BB * HH;         // state after GRU
  float* vpt  = ws + o; o += NPAD;                    // V_q part (row 0)
  float* sp1  = ws + o; o += (size_t)BB * NPAD;
  float* sp2  = ws + o; o += (size_t)BB * NPAD;

  const int Mq = TQ * BB;        // 4096
  const int Mp = TP * BB;        // 131072
  const int ldW = 2 * HH;        // 512

  // 1) Question token part: Yq = question @ W1_q[:, :H].T
  gemm_xwT_kernel<<<(Mq + 255) / 256, 256, 0, stream>>>(
      question, W1_q, Yq, Mq, HH, HH, ldW, AA);
  // 2) V_q part (single row): vpt = V_q @ W1_q[:, H:].T
  gemm_xwT_kernel<<<1, 256, 0, stream>>>(
      V_q, W1_q + HH, vpt, 1, HH, HH, ldW, AA);
  // 3) Question masked logits.
  logits_mask_kernel<<<(Mq + 255) / 256, 256, 0, stream>>>(
      Yq, vpt, 0, W2_q, qmask, TQ, qlog, nullptr);
  // 4) Question softmax + pool -> state.
  softmax_pool_kernel<<<BB, 256, 0, stream>>>(qlog, question, TQ, st1);

  // 5) Passage token part (shared by both passes): Pp = passage @ W1_p[:, :H].T
  gemm_xwT_kernel<<<(Mp + 255) / 256, 256, 0, stream>>>(
      passage, W1_p, Pp, Mp, HH, HH, ldW, AA);

  // 6) Start pointer: state part + logits + pool.
  gemm_xwT_kernel<<<1, 256, 0, stream>>>(
      st1, W1_p + HH, sp1, BB, HH, HH, ldW, AA);
  logits_mask_kernel<<<(Mp + 255) / 256, 256, 0, stream>>>(
      Pp, sp1, NPAD, W2_p, pmask, TP, plog, out);               // ans_start [B,Tp]
  softmax_pool_kernel<<<BB, 256, 0, stream>>>(plog, passage, TP, ci);

  // 7) GRU step.
  gru_kernel<<<BB, 256, 0, stream>>>(ci, st1, W_ih, W_hh, b_ih, b_hh, st2);

  // 8) End pointer.
  gemm_xwT_kernel<<<1, 256, 0, stream>>>(
      st2, W1_p + HH, sp2, BB, HH, HH, ldW, AA);
  logits_mask_kernel<<<(Mp + 255) / 256, 256, 0, stream>>>(
      Pp, sp2, NPAD, W2_p, pmask, TP, plog, out + (size_t)BB * TP); // ans_end
}